// CapsuleLayer_45148696216021
// MI455X (gfx1250) — compile-verified
//
#include <hip/hip_runtime.h>

typedef __attribute__((ext_vector_type(2))) float v2f;
typedef __attribute__((ext_vector_type(8))) float v8f;

#define B_SZ  128
#define I_SZ  1024
#define O_SZ  1024
#define N_REL 128

#define KC      32            // K-chunk staged in LDS per iteration
#define LDS_LW  144           // padded row stride (floats); 144%64=16 banks

__device__ __forceinline__ v8f wmma_f32_16x16x4(v2f a, v2f b, v8f c) {
  // 8 args: (neg_a, A, neg_b, B, c_mod, C, reuse_a, reuse_b)
  return __builtin_amdgcn_wmma_f32_16x16x4_f32(false, a, false, b, (short)0, c,
                                               false, false);
}

// Issue one async 16B global->LDS copy (ASYNCcnt-tracked, no VGPR round trip).
// The LDS offset is derived from the shared-array pointer (ptrtoint) so the
// object escapes into the asm and the "memory" clobber legally covers it --
// otherwise LLVM deletes the LDS reads as loads of a never-written local.
__device__ __forceinline__ void async_g2l_b128(float* lds_dst,
                                               const float* gptr) {
  // generic ptr to LDS = {SHARED_BASE[63:32], lds_offset[31:0]}
  const unsigned lds_byte_off = (unsigned)(size_t)(void*)lds_dst;
  asm volatile("global_load_async_to_lds_b128 %0, %1, off"
               :: "v"(lds_byte_off), "v"(gptr)
               : "memory");
}

__device__ __forceinline__ void wait_async_zero() {
  asm volatile("s_wait_asynccnt 0x0" ::: "memory");
}

// ---------------------------------------------------------------------------
// Phase 1: caps[b,n,o] = sum_i x[b,i] * rel_W[n,i,o] + rel_b[n,o]
// Stored TRANSPOSED per batch: capsT[b][o][n] (realizes the torch
// reshape/transpose so phase 2 reads a plain row-major [128][1024]).
//
// rel_W (512 MB) streams exactly once -> stage 32x128 K-chunks of it in LDS
// via async b128 copies, double buffered so DMA of chunk k+1 overlaps the
// 8 WMMAs of chunk k. grid (O/128, B/16, N), block 256 (8 waves).
// ---------------------------------------------------------------------------
__global__ void __launch_bounds__(256)
caps_gemm_kernel(const float* __restrict__ x, const float* __restrict__ rel_W,
                 const float* __restrict__ rel_b, float* __restrict__ capsT) {
  __shared__ float ldsB[2][KC * LDS_LW];

  const int t     = threadIdx.x;
  const int lane  = t & 31;
  const int wave  = t >> 5;
  const int n     = blockIdx.z;
  const int brow0 = blockIdx.y * 16;
  const int oblk  = blockIdx.x * 128;
  const int owave = wave * 16;
  const int lm = lane & 15;
  const int kh = (lane >> 4) << 1;

  const float* __restrict__ W    = rel_W + (size_t)n * (I_SZ * (size_t)O_SZ);
  const float* __restrict__ xrow = x + (size_t)(brow0 + lm) * I_SZ;

  // Loader mapping: 32 rows x 128 cols = 1024 float4; 4 per thread.
  // flat f = t + 256*q -> row = f>>5, col4 = f&31.
  int lrow[4], lcol4[4];
#pragma unroll
  for (int q = 0; q < 4; ++q) {
    const int f = t + (q << 8);
    lrow[q]  = f >> 5;
    lcol4[q] = f & 31;
  }

  // Preload chunk 0 into buffer 0.
#pragma unroll
  for (int q = 0; q < 4; ++q) {
    async_g2l_b128(&ldsB[0][lrow[q] * LDS_LW + (lcol4[q] << 2)],
                   W + (size_t)lrow[q] * O_SZ + oblk + (lcol4[q] << 2));
  }

  v8f acc = {};
  int buf = 0;
  for (int k0 = 0; k0 < I_SZ; k0 += KC) {
    wait_async_zero();     // own async writes to LDS complete
    __syncthreads();       // everyone's writes visible; prev buffer free

    if (k0 + KC < I_SZ) {  // kick off DMA of the next chunk into other buffer
#pragma unroll
      for (int q = 0; q < 4; ++q) {
        async_g2l_b128(&ldsB[buf ^ 1][lrow[q] * LDS_LW + (lcol4[q] << 2)],
                       W + (size_t)(k0 + KC + lrow[q]) * O_SZ + oblk +
                           (lcol4[q] << 2));
      }
    }

    // 8 WMMA steps out of the resident LDS chunk.
#pragma unroll
    for (int ks = 0; ks < KC; ks += 4) {
      v2f a = *(const v2f*)(xrow + k0 + ks + kh);        // A: x row, K-contig
      v2f b;
      b.x = ldsB[buf][(ks + kh) * LDS_LW + owave + lm];
      b.y = ldsB[buf][(ks + kh + 1) * LDS_LW + owave + lm];
      acc = wmma_f32_16x16x4(a, b, acc);
    }
    buf ^= 1;
  }

  const int   o     = oblk + owave + lm;
  const float bias  = rel_b[(size_t)n * O_SZ + o];
  const int   rbase = brow0 + ((lane >> 4) << 3);
#pragma unroll
  for (int r = 0; r < 8; ++r)
    capsT[((size_t)(rbase + r) * O_SZ + o) * N_REL + n] = acc[r] + bias;
}

// ---------------------------------------------------------------------------
// Phase 2: u_hat[b,i,n] = sum_o rw[b,i,o] * capsT[b][n][o]   (NT GEMM)
// rw streams; capsT[b] slice (512 KB) is L2-resident. grid (I/128, N/16, B).
// ---------------------------------------------------------------------------
__global__ void __launch_bounds__(256)
uhat_gemm_kernel(const float* __restrict__ rw, const float* __restrict__ capsT,
                 float* __restrict__ uhat) {
  const int lane  = threadIdx.x & 31;
  const int wave  = threadIdx.x >> 5;
  const int bb    = blockIdx.z;
  const int irow0 = blockIdx.x * 128 + wave * 16;
  const int ncol0 = blockIdx.y * 16;
  const int lm = lane & 15;
  const int kh = (lane >> 4) << 1;

  const float* __restrict__ A  = rw + (size_t)bb * (I_SZ * (size_t)O_SZ) +
                                 (size_t)(irow0 + lm) * O_SZ;
  const float* __restrict__ Bt = capsT + (size_t)bb * (N_REL * (size_t)O_SZ) +
                                 (size_t)(ncol0 + lm) * O_SZ;

  v8f acc = {};
#pragma unroll 8
  for (int k0 = 0; k0 < O_SZ; k0 += 4) {
    v2f a = *(const v2f*)(A + k0 + kh);
    v2f b = *(const v2f*)(Bt + k0 + kh);
    __builtin_prefetch(A + k0 + kh + 64, 0, 3);
    acc = wmma_f32_16x16x4(a, b, acc);
  }

  const int col   = ncol0 + lm;
  const int rbase = irow0 + ((lane >> 4) << 3);
#pragma unroll
  for (int r = 0; r < 8; ++r)
    uhat[((size_t)bb * I_SZ + rbase + r) * N_REL + col] = acc[r];
}

// ---------------------------------------------------------------------------
__global__ void __launch_bounds__(256) zero_kernel(float* __restrict__ p, int n4) {
  int idx = blockIdx.x * blockDim.x + threadIdx.x;
  if (idx < n4) ((float4*)p)[idx] = make_float4(0.f, 0.f, 0.f, 0.f);
}

// Per-(b,i) softmax stats over the capsule axis j (stride I_SZ).
// grid (B*I/256), block 256 -- consecutive i => coalesced.
__global__ void __launch_bounds__(256)
softmax_stats_kernel(const float* __restrict__ blog, float* __restrict__ smax,
                     float* __restrict__ sden) {
  const int idx = blockIdx.x * 256 + threadIdx.x;   // = b*I + i
  const int bb  = idx >> 10;
  const int i   = idx & (I_SZ - 1);
  const float* p = blog + (size_t)bb * (N_REL * (size_t)I_SZ) + i;
  float m = -3.402823466e38f;
  for (int j = 0; j < N_REL; ++j) m = fmaxf(m, p[(size_t)j * I_SZ]);
  float d = 0.f;
  for (int j = 0; j < N_REL; ++j) d += __expf(p[(size_t)j * I_SZ] - m);
  smax[idx] = m;
  sden[idx] = d;
}

// ---------------------------------------------------------------------------
// Phase 3: s[b,j,k] = sum_i softmax(blog)[b,j,i] * u_hat[b,i,k]
// Softmax fused into the A-fragment (c = exp(bl-m)/d). NN GEMM, K=1024.
// grid (N/16, B), block 256; wave w owns j-tile w.
// ---------------------------------------------------------------------------
__global__ void __launch_bounds__(256)
s_gemm_kernel(const float* __restrict__ blog, const float* __restrict__ smax,
              const float* __restrict__ sden, const float* __restrict__ uhat,
              float* __restrict__ s) {
  const int lane  = threadIdx.x & 31;
  const int wave  = threadIdx.x >> 5;
  const int bb    = blockIdx.y;
  const int jrow0 = wave * 16;
  const int kcol0 = blockIdx.x * 16;
  const int lm = lane & 15;
  const int kh = (lane >> 4) << 1;

  const float* __restrict__ blrow = blog + (size_t)bb * (N_REL * (size_t)I_SZ) +
                                    (size_t)(jrow0 + lm) * I_SZ;
  const float* __restrict__ mx = smax + (size_t)bb * I_SZ;
  const float* __restrict__ dn = sden + (size_t)bb * I_SZ;
  const float* __restrict__ U  = uhat + (size_t)bb * (I_SZ * (size_t)N_REL) +
                                 kcol0 + lm;

  v8f acc = {};
#pragma unroll 4
  for (int k0 = 0; k0 < I_SZ; k0 += 4) {
    const int kb = k0 + kh;
    v2f bl = *(const v2f*)(blrow + kb);
    v2f m2 = *(const v2f*)(mx + kb);
    v2f d2 = *(const v2f*)(dn + kb);
    v2f a;
    a.x = __expf(bl.x - m2.x) / d2.x;
    a.y = __expf(bl.y - m2.y) / d2.y;
    v2f b;
    b.x = U[(size_t)kb * N_REL];
    b.y = U[(size_t)(kb + 1) * N_REL];
    acc = wmma_f32_16x16x4(a, b, acc);
  }

  const int col   = kcol0 + lm;
  const int rbase = jrow0 + ((lane >> 4) << 3);
#pragma unroll
  for (int r = 0; r < 8; ++r)
    s[((size_t)bb * N_REL + rbase + r) * N_REL + col] = acc[r];
}

// squash along last axis: v = s * sn / ((1+sn)*sqrt(sn)); one thread per row.
__global__ void __launch_bounds__(256)
squash_kernel(const float* __restrict__ s, float* __restrict__ v) {
  const int idx = blockIdx.x * 256 + threadIdx.x;   // = b*N + j
  const float* row = s + (size_t)idx * N_REL;
  float sn = 0.f;
  for (int k = 0; k < N_REL; ++k) { float t = row[k]; sn += t * t; }
  const float scale = sn / ((1.0f + sn) * sqrtf(sn));
  float* orow = v + (size_t)idx * N_REL;
  for (int k = 0; k < N_REL; ++k) orow[k] = row[k] * scale;
}

// ---------------------------------------------------------------------------
// Phase 4: blog[b,j,i] += sum_k v[b,j,k] * u_hat[b,i,k]   (NT GEMM, K=128)
// grid (I/128, N/16, B), block 256
// ---------------------------------------------------------------------------
__global__ void __launch_bounds__(256)
bupdate_gemm_kernel(const float* __restrict__ v, const float* __restrict__ uhat,
                    float* __restrict__ blog) {
  const int lane  = threadIdx.x & 31;
  const int wave  = threadIdx.x >> 5;
  const int bb    = blockIdx.z;
  const int jrow0 = blockIdx.y * 16;
  const int icol0 = blockIdx.x * 128 + wave * 16;
  const int lm = lane & 15;
  const int kh = (lane >> 4) << 1;

  const float* __restrict__ A  = v + ((size_t)bb * N_REL + jrow0 + lm) * N_REL;
  const float* __restrict__ Bt = uhat + ((size_t)bb * I_SZ + icol0 + lm) * N_REL;

  v8f acc = {};
#pragma unroll
  for (int k0 = 0; k0 < N_REL; k0 += 4) {
    v2f a = *(const v2f*)(A + k0 + kh);
    v2f b = *(const v2f*)(Bt + k0 + kh);
    acc = wmma_f32_16x16x4(a, b, acc);
  }

  const int col   = icol0 + lm;
  const int rbase = jrow0 + ((lane >> 4) << 3);
#pragma unroll
  for (int r = 0; r < 8; ++r) {
    float* p = blog + ((size_t)bb * N_REL + rbase + r) * I_SZ + col;
    *p += acc[r];
  }
}

// ---------------------------------------------------------------------------
extern "C" void kernel_launch(void* const* d_in, const int* in_sizes, int n_in,
                              void* d_out, int out_size, void* d_ws, size_t ws_size,
                              hipStream_t stream) {
  (void)in_sizes; (void)n_in; (void)out_size; (void)ws_size;
  const float* x     = (const float*)d_in[0];
  // d_in[1] = edge_index, d_in[2] = edge_type: unused by the reference math
  const float* rel_W = (const float*)d_in[3];
  const float* rel_b = (const float*)d_in[4];
  const float* rw    = (const float*)d_in[5];   // route_weights[...,0], [N][I][O]
  float* out = (float*)d_out;

  float* ws    = (float*)d_ws;
  float* capsT = ws;                       // 16,777,216 f32 (64 MB)
  float* blog  = ws;                       // aliases capsT after phase 2
  float* uhat  = ws + 16777216;            // 16,777,216 f32 (64 MB)
  float* smax  = ws + 33554432;            // 131,072 f32
  float* sden  = smax + 131072;            // 131,072 f32
  float* sv    = sden + 131072;            // 2,097,152 f32 (s and v, in place)

  const dim3 blk(256);
  caps_gemm_kernel<<<dim3(8, 8, 128), blk, 0, stream>>>(x, rel_W, rel_b, capsT);
  uhat_gemm_kernel<<<dim3(8, 8, 128), blk, 0, stream>>>(rw, capsT, uhat);
  zero_kernel<<<dim3(16384), blk, 0, stream>>>(blog, 4194304);

  for (int it = 0; it < 3; ++it) {
    softmax_stats_kernel<<<dim3(512), blk, 0, stream>>>(blog, smax, sden);
    s_gemm_kernel<<<dim3(8, 128), blk, 0, stream>>>(blog, smax, sden, uhat, sv);
    float* vout = (it == 2) ? out : sv;
    squash_kernel<<<dim3(64), blk, 0, stream>>>(sv, vout);
    if (it < 2)
      bupdate_gemm_kernel<<<dim3(8, 8, 128), blk, 0, stream>>>(sv, uhat, blog);
  }
}